// Cell_50096498541139
// MI455X (gfx1250) — compile-verified
//
#include <hip/hip_runtime.h>
#include <math.h>

typedef __attribute__((ext_vector_type(2))) float v2f;
typedef __attribute__((ext_vector_type(8))) float v8f;

#define QDIM 256          // 2^8 Hilbert dim
#define BATCH 64
#define SEQLEN 8
#define NWIRES 8
#define NCLASS 10
#define LDSP 17           // padded LDS row stride (floats) -> conflict-free banks

// ---------------------------------------------------------------------------
// Compose the full per-step circuit unitary G (256x256 complex, split re/im).
// Gates are batch/time invariant, so this runs once. Single workgroup.
// ---------------------------------------------------------------------------
__global__ __launch_bounds__(256) void build_G(const float* __restrict__ qrot,
                                               const float* __restrict__ qent,
                                               float* __restrict__ Gr,
                                               float* __restrict__ Gi)
{
    const int t = threadIdx.x;
    for (int idx = t; idx < QDIM * QDIM; idx += 256) {
        int r = idx >> 8, c = idx & 255;
        Gr[idx] = (r == c) ? 1.f : 0.f;
        Gi[idx] = 0.f;
    }
    __syncthreads();

    for (int l = 0; l < 2; ++l) {
        // ---- single-qubit U3 on every wire (wire w = bit (7-w), MSB first) ----
        for (int q = 0; q < NWIRES; ++q) {
            const float* p = qrot + (l * NWIRES + q) * 3;
            float th = p[0], ph = p[1], lam = p[2];
            float ch = cosf(0.5f * th), sh = sinf(0.5f * th);
            float u00r = ch,                u00i = 0.f;
            float u01r = -cosf(lam) * sh,   u01i = -sinf(lam) * sh;
            float u10r =  cosf(ph) * sh,    u10i =  sinf(ph) * sh;
            float u11r =  cosf(ph + lam) * ch, u11i = sinf(ph + lam) * ch;
            int maskT = 1 << (7 - q);
            int low = maskT - 1;
            for (int idx = t; idx < (QDIM / 2) * QDIM; idx += 256) {
                int col = idx & 255;
                int pr  = idx >> 8;                       // 0..127 row-pair id
                int r0 = ((pr & ~low) << 1) | (pr & low); // insert 0 at bit
                int r1 = r0 | maskT;
                float ar = Gr[r0 * QDIM + col], ai = Gi[r0 * QDIM + col];
                float br = Gr[r1 * QDIM + col], bi = Gi[r1 * QDIM + col];
                Gr[r0 * QDIM + col] = u00r * ar - u00i * ai + u01r * br - u01i * bi;
                Gi[r0 * QDIM + col] = u00r * ai + u00i * ar + u01r * bi + u01i * br;
                Gr[r1 * QDIM + col] = u10r * ar - u10i * ai + u11r * br - u11i * bi;
                Gi[r1 * QDIM + col] = u10r * ai + u10i * ar + u11r * bi + u11i * br;
            }
            __syncthreads();
        }
        // ---- controlled-U3: control q, target (q+1)%8 ----
        for (int q = 0; q < NWIRES; ++q) {
            const float* p = qent + (l * NWIRES + q) * 3;
            float th = p[0], ph = p[1], lam = p[2];
            float ch = cosf(0.5f * th), sh = sinf(0.5f * th);
            float u00r = ch,                u00i = 0.f;
            float u01r = -cosf(lam) * sh,   u01i = -sinf(lam) * sh;
            float u10r =  cosf(ph) * sh,    u10i =  sinf(ph) * sh;
            float u11r =  cosf(ph + lam) * ch, u11i = sinf(ph + lam) * ch;
            int tq = (q + 1) & 7;
            int maskC = 1 << (7 - q);
            int maskT = 1 << (7 - tq);
            int low = maskT - 1;
            for (int idx = t; idx < (QDIM / 2) * QDIM; idx += 256) {
                int col = idx & 255;
                int pr  = idx >> 8;
                int r0 = ((pr & ~low) << 1) | (pr & low);
                if (!(r0 & maskC)) continue;              // control bit must be 1
                int r1 = r0 | maskT;
                float ar = Gr[r0 * QDIM + col], ai = Gi[r0 * QDIM + col];
                float br = Gr[r1 * QDIM + col], bi = Gi[r1 * QDIM + col];
                Gr[r0 * QDIM + col] = u00r * ar - u00i * ai + u01r * br - u01i * bi;
                Gi[r0 * QDIM + col] = u00r * ai + u00i * ar + u01r * bi + u01i * br;
                Gr[r1 * QDIM + col] = u10r * ar - u10i * ai + u11r * br - u11i * bi;
                Gi[r1 * QDIM + col] = u10r * ai + u10i * ar + u11r * bi + u11i * br;
            }
            __syncthreads();
        }
    }
}

// ---------------------------------------------------------------------------
// Partial trace over wires 0..3 (high index nibble) + re-encode features x[:,t,:].
// rho_new[j*16+k][j2*16+l] = red[j][j2] * feat[k][l]. One block per batch.
// ---------------------------------------------------------------------------
__global__ __launch_bounds__(256) void reencode(float* __restrict__ rho_r,
                                                float* __restrict__ rho_i,
                                                const float* __restrict__ x,
                                                int tstep, int first)
{
    __shared__ float red_r[256], red_i[256], fmat[256];
    const int b = blockIdx.x;
    const int tid = threadIdx.x;
    float* Rr = rho_r + (size_t)b * QDIM * QDIM;
    float* Ri = rho_i + (size_t)b * QDIM * QDIM;
    const int j = tid >> 4, k = tid & 15;

    if (first) {                                // rho0 = |0><0|  -> red = |0><0|
        red_r[tid] = (tid == 0) ? 1.f : 0.f;
        red_i[tid] = 0.f;
    } else {
        float sr = 0.f, si = 0.f;
        for (int i = 0; i < 16; ++i) {
            int row = i * 16 + j, col = i * 16 + k;
            sr += Rr[row * QDIM + col];
            si += Ri[row * QDIM + col];
        }
        red_r[tid] = sr; red_i[tid] = si;
    }
    // feature density matrix (real symmetric 16x16): product state of 4 qubits
    {
        float v = 1.f;
        for (int f = 0; f < 4; ++f) {
            float xv = x[b * SEQLEN * 4 + tstep * 4 + f];
            float s = sqrtf(fmaxf((1.f - xv) * xv, 0.f));
            int kb = (j >> f) & 1, lb = (k >> f) & 1;
            v *= kb ? (lb ? xv : s) : (lb ? s : (1.f - xv));
        }
        fmat[tid] = v;
    }
    __syncthreads();

    const int row = tid, rj = row >> 4, rk = row & 15;
    for (int col = 0; col < QDIM; ++col) {
        int cj = col >> 4, cl = col & 15;
        float f = fmat[rk * 16 + cl];
        Rr[row * QDIM + col] = red_r[rj * 16 + cj] * f;
        Ri[row * QDIM + col] = red_i[rj * 16 + cj] * f;
    }
}

// ---------------------------------------------------------------------------
// T = G @ rho   (complex fp32 GEMM via V_WMMA_F32_16X16X4_F32).
// Block = 8 wave32s covering 128 rows of one 16-col stripe. The 256x16
// complex B-stripe of rho is staged once into LDS (coalesced), then all 8
// waves feed WMMA fragments from LDS (bank-conflict-free, stride 17).
// Grid (2, 16, BATCH).
// ---------------------------------------------------------------------------
__global__ __launch_bounds__(256) void gemm_left(const float* __restrict__ Gr,
                                                 const float* __restrict__ Gi,
                                                 const float* __restrict__ rho_r,
                                                 const float* __restrict__ rho_i,
                                                 float* __restrict__ out_r,
                                                 float* __restrict__ out_i)
{
    __shared__ float LBr[QDIM * LDSP];
    __shared__ float LBi[QDIM * LDSP];
    const int tid  = threadIdx.x;
    const int wave = tid >> 5, lane = tid & 31;
    const int half = lane >> 4, l15 = lane & 15;
    const int col0 = blockIdx.y * 16;
    const int row0 = (blockIdx.x * 8 + wave) * 16;
    const size_t base = (size_t)blockIdx.z * QDIM * QDIM;
    const float* Rr = rho_r + base;
    const float* Ri = rho_i + base;
    float* Tr = out_r + base;
    float* Ti = out_i + base;

    // ---- stage B stripe: LB[k][n] = rho[k][col0+n] ----
    for (int c = 0; c < 16; ++c) {
        int idx = c * 256 + tid;                   // 0..4095
        int k = idx >> 4, n = idx & 15;
        LBr[k * LDSP + n] = Rr[k * QDIM + col0 + n];
        LBi[k * LDSP + n] = Ri[k * QDIM + col0 + n];
    }
    __syncthreads();

    const int m = row0 + l15;                      // A row for this lane
    v8f cre = {}; v8f cim = {};
    for (int k0 = 0; k0 < QDIM; k0 += 4) {
        const int kk = k0 + 2 * half;
        v2f ar = *(const v2f*)(Gr + m * QDIM + kk);
        v2f ai = *(const v2f*)(Gi + m * QDIM + kk);
        v2f br; br.x = LBr[kk * LDSP + l15]; br.y = LBr[(kk + 1) * LDSP + l15];
        v2f bi; bi.x = LBi[kk * LDSP + l15]; bi.y = LBi[(kk + 1) * LDSP + l15];
        v2f an = -ai;   // fp32 WMMA has no A-neg modifier -> negate fragment
        cre = __builtin_amdgcn_wmma_f32_16x16x4_f32(false, ar, false, br, (short)0, cre, false, false);
        cre = __builtin_amdgcn_wmma_f32_16x16x4_f32(false, an, false, bi, (short)0, cre, false, false);
        cim = __builtin_amdgcn_wmma_f32_16x16x4_f32(false, ar, false, bi, (short)0, cim, false, false);
        cim = __builtin_amdgcn_wmma_f32_16x16x4_f32(false, ai, false, br, (short)0, cim, false, false);
    }
    for (int v = 0; v < 8; ++v) {
        int mm = row0 + v + 8 * half, nn = col0 + l15;
        Tr[mm * QDIM + nn] = cre[v];
        Ti[mm * QDIM + nn] = cim[v];
    }
}

// ---------------------------------------------------------------------------
// rho = (T @ G^dagger) * deph. The conjugate-transposed G stripe
// B[k][n] = conj(G[col0+n][k]) is staged into LDS with the conjugation folded
// into the (perfectly coalesced) staging loads. Dephasing fused via popcount.
// ---------------------------------------------------------------------------
__global__ __launch_bounds__(256) void gemm_right_deph(const float* __restrict__ Tr_,
                                                       const float* __restrict__ Ti_,
                                                       const float* __restrict__ Gr,
                                                       const float* __restrict__ Gi,
                                                       const float* __restrict__ g_raw,
                                                       float* __restrict__ rho_r,
                                                       float* __restrict__ rho_i)
{
    __shared__ float LBr[QDIM * LDSP];
    __shared__ float LBi[QDIM * LDSP];
    const int tid  = threadIdx.x;
    const int wave = tid >> 5, lane = tid & 31;
    const int half = lane >> 4, l15 = lane & 15;
    const int col0 = blockIdx.y * 16;
    const int row0 = (blockIdx.x * 8 + wave) * 16;
    const size_t base = (size_t)blockIdx.z * QDIM * QDIM;
    const float* Tr = Tr_ + base;
    const float* Ti = Ti_ + base;
    float* Rr = rho_r + base;
    float* Ri = rho_i + base;

    // ---- stage conj(G) stripe: LB[k][n] = conj(G[col0+n][k]) ----
    for (int c = 0; c < 16; ++c) {
        int k = tid;                               // 0..255, coalesced in k
        LBr[k * LDSP + c] =  Gr[(col0 + c) * QDIM + k];
        LBi[k * LDSP + c] = -Gi[(col0 + c) * QDIM + k];
    }
    __syncthreads();

    const int m = row0 + l15;
    v8f cre = {}; v8f cim = {};
    for (int k0 = 0; k0 < QDIM; k0 += 4) {
        const int kk = k0 + 2 * half;
        v2f ar = *(const v2f*)(Tr + m * QDIM + kk);
        v2f ai = *(const v2f*)(Ti + m * QDIM + kk);
        v2f br; br.x = LBr[kk * LDSP + l15]; br.y = LBr[(kk + 1) * LDSP + l15];
        v2f bi; bi.x = LBi[kk * LDSP + l15]; bi.y = LBi[(kk + 1) * LDSP + l15];
        v2f an = -ai;
        cre = __builtin_amdgcn_wmma_f32_16x16x4_f32(false, ar, false, br, (short)0, cre, false, false);
        cre = __builtin_amdgcn_wmma_f32_16x16x4_f32(false, an, false, bi, (short)0, cre, false, false);
        cim = __builtin_amdgcn_wmma_f32_16x16x4_f32(false, ar, false, bi, (short)0, cim, false, false);
        cim = __builtin_amdgcn_wmma_f32_16x16x4_f32(false, ai, false, br, (short)0, cim, false, false);
    }
    const float g = log1pf(expf(g_raw[0]));        // softplus
    const float c2 = -0.5f * g * g;
    for (int v = 0; v < 8; ++v) {
        int mm = row0 + v + 8 * half, nn = col0 + l15;
        float f = expf(c2 * (float)__popc(mm ^ nn));
        Rr[mm * QDIM + nn] = cre[v] * f;
        Ri[mm * QDIM + nn] = cim[v] * f;
    }
}

// ---------------------------------------------------------------------------
// Z-expectations from Re(diag(rho)); feats[b][t*8+i].
// ---------------------------------------------------------------------------
__global__ __launch_bounds__(256) void expectations(const float* __restrict__ rho_r,
                                                    float* __restrict__ feats,
                                                    int tstep)
{
    __shared__ float sd[QDIM];
    const int b = blockIdx.x, a = threadIdx.x;
    sd[a] = rho_r[(size_t)b * QDIM * QDIM + (size_t)a * QDIM + a];
    __syncthreads();
    if (a < NWIRES) {
        const int shift = 7 - a;
        float s = 0.f;
        for (int d = 0; d < QDIM; ++d)
            s += ((d >> shift) & 1) ? -sd[d] : sd[d];
        feats[b * (SEQLEN * NWIRES) + tstep * NWIRES + a] = s;
    }
}

// ---------------------------------------------------------------------------
// Final FC: out[b][c] = feats[b] . fc_w[c] + fc_b[c]
// ---------------------------------------------------------------------------
__global__ __launch_bounds__(256) void fc_kernel(const float* __restrict__ feats,
                                                 const float* __restrict__ w,
                                                 const float* __restrict__ bias,
                                                 float* __restrict__ out)
{
    int idx = blockIdx.x * blockDim.x + threadIdx.x;
    if (idx >= BATCH * NCLASS) return;
    int b = idx / NCLASS, c = idx % NCLASS;
    float s = bias[c];
    for (int k = 0; k < SEQLEN * NWIRES; ++k)
        s += feats[b * 64 + k] * w[c * 64 + k];
    out[idx] = s;
}

extern "C" void kernel_launch(void* const* d_in, const int* in_sizes, int n_in,
                              void* d_out, int out_size, void* d_ws, size_t ws_size,
                              hipStream_t stream)
{
    const float* x    = (const float*)d_in[0];   // (64,8,4)
    const float* qrot = (const float*)d_in[1];   // (2,8,3)
    const float* qent = (const float*)d_in[2];   // (2,8,3)
    const float* graw = (const float*)d_in[3];   // scalar
    const float* fcw  = (const float*)d_in[4];   // (10,64)
    const float* fcb  = (const float*)d_in[5];   // (10,)

    float* ws    = (float*)d_ws;
    float* Gr    = ws;
    float* Gi    = Gr + QDIM * QDIM;
    float* rho_r = Gi + QDIM * QDIM;
    float* rho_i = rho_r + (size_t)BATCH * QDIM * QDIM;
    float* Tr    = rho_i + (size_t)BATCH * QDIM * QDIM;
    float* Ti    = Tr + (size_t)BATCH * QDIM * QDIM;
    float* feats = Ti + (size_t)BATCH * QDIM * QDIM;

    build_G<<<1, 256, 0, stream>>>(qrot, qent, Gr, Gi);

    dim3 gg(QDIM / (16 * 8), QDIM / 16, BATCH);  // (2, 16, 64) blocks of 8 waves
    for (int t = 0; t < SEQLEN; ++t) {
        reencode<<<BATCH, 256, 0, stream>>>(rho_r, rho_i, x, t, (t == 0) ? 1 : 0);
        gemm_left<<<gg, 256, 0, stream>>>(Gr, Gi, rho_r, rho_i, Tr, Ti);
        gemm_right_deph<<<gg, 256, 0, stream>>>(Tr, Ti, Gr, Gi, graw, rho_r, rho_i);
        expectations<<<BATCH, 256, 0, stream>>>(rho_r, feats, t);
    }
    fc_kernel<<<(BATCH * NCLASS + 255) / 256, 256, 0, stream>>>(feats, fcw, fcb, (float*)d_out);
}